// GATModel_53231824666770
// MI455X (gfx1250) — compile-verified
//
#include <hip/hip_runtime.h>
#include <math.h>

typedef float v2f __attribute__((ext_vector_type(2)));
typedef float v8f __attribute__((ext_vector_type(8)));

#define NEG_SLOPE 0.2f
#define EPSF      1e-16f

// ---------------------------------------------------------------------------
// Dense GEMM C[N,M] = A[N,K] * B[K,M] using V_WMMA_F32_16X16X4_F32.
// One wave32 computes one 16x16 tile of C. f32 in / f32 accumulate => exact
// precision parity with the f32 reference. Requires N%16==0, M%16==0, K%4==0.
// A-operand layout (ISA 7.12.2, 32-bit A 16x4): lanes 0-15 hold K=0,1 for
// M=lane; lanes 16-31 hold K=2,3 -> one contiguous v2f load per lane.
// B-operand (4x16, row striped across lanes): v0 = row k0(+2), v1 = row k0+1(+3).
// ---------------------------------------------------------------------------
__global__ void gemm16_wmma_f32(const float* __restrict__ A,
                                const float* __restrict__ B,
                                float* __restrict__ C,
                                int N, int K, int M) {
  const int lane          = threadIdx.x & 31;
  const int waveInBlock   = threadIdx.x >> 5;
  const int wavesPerBlock = blockDim.x >> 5;
  const int tile          = blockIdx.x * wavesPerBlock + waveInBlock;
  const int tilesM        = M >> 4;
  const int tilesN        = N >> 4;
  if (tile >= tilesN * tilesM) return;   // wave-uniform guard (EXEC stays ~0xffffffff)
  const int tm   = tile / tilesM;
  const int tn   = tile % tilesM;
  const int half = lane >> 4;            // 0: K pair {0,1}; 1: K pair {2,3}
  const int l15  = lane & 15;

  const float* arow = A + (size_t)(tm * 16 + l15) * K + 2 * half;
  const float* bcol = B + (size_t)(tn * 16 + l15);

  v8f acc = {};
  for (int k0 = 0; k0 < K; k0 += 4) {
    __builtin_prefetch(arow + k0 + 16, 0, 3);        // global_prefetch_b8
    v2f a = *(const v2f*)(arow + k0);                // A[m][k0+2*half .. +1]
    const int kr = k0 + 2 * half;
    v2f b;
    b.x = bcol[(size_t)kr * M];                      // B[kr][n]
    b.y = bcol[(size_t)(kr + 1) * M];                // B[kr+1][n]
    acc = __builtin_amdgcn_wmma_f32_16x16x4_f32(
        /*neg_a=*/false, a, /*neg_b=*/false, b,
        /*c_mod=*/(short)0, acc, /*reuse_a=*/false, /*reuse_b=*/false);
  }

  // C/D layout: VGPR r -> row tm*16 + r + 8*half, col tn*16 + (lane&15)
  float* crow = C + (size_t)(tm * 16 + half * 8) * M + tn * 16 + l15;
#pragma unroll
  for (int r = 0; r < 8; ++r) crow[(size_t)r * M] = acc[r];
}

// ---------------------------------------------------------------------------
// Utility: fill buffer with a constant (grid-stride).
// ---------------------------------------------------------------------------
__global__ void fill_f32(float* __restrict__ p, float v, long long n) {
  long long i      = (long long)blockIdx.x * blockDim.x + threadIdx.x;
  long long stride = (long long)gridDim.x * blockDim.x;
  for (; i < n; i += stride) p[i] = v;
}

// ---------------------------------------------------------------------------
// Per-(node,head) attention logits: a_src[n,h] = <h[n,h,:], att_src[h,:]>, same
// for a_dst.  h is [N, H*C] row-major, thread i = n*H + head.
// ---------------------------------------------------------------------------
__global__ void att_scores(const float* __restrict__ h,
                           const float* __restrict__ att_s,
                           const float* __restrict__ att_d,
                           float* __restrict__ a_src,
                           float* __restrict__ a_dst,
                           int N, int H, int C) {
  int i = blockIdx.x * blockDim.x + threadIdx.x;
  if (i >= N * H) return;
  int head = i % H;
  const float* hp = h + (size_t)i * C;
  const float* ws = att_s + (size_t)head * C;
  const float* wd = att_d + (size_t)head * C;
  float ss = 0.f, sd = 0.f;
  for (int c = 0; c < C; ++c) {
    float v = hp[c];
    ss += v * ws[c];
    sd += v * wd[c];
  }
  a_src[i] = ss;
  a_dst[i] = sd;
}

// Edge helpers. edge_index is [2, E0] (row0 = src, row1 = dst); edges
// E0..E0+N-1 are implicit self loops (i,i).
__device__ __forceinline__ int edge_src(const int* ei, int E0, int e) {
  return (e < E0) ? ei[e] : (e - E0);
}
__device__ __forceinline__ int edge_dst(const int* ei, int E0, int e) {
  return (e < E0) ? ei[E0 + e] : (e - E0);
}
__device__ __forceinline__ float leaky(float a) {
  return a > 0.f ? a : a * NEG_SLOPE;
}

__device__ __forceinline__ float atomicMaxF32(float* addr, float val) {
  unsigned int* ua = (unsigned int*)addr;
  unsigned int old = *ua, assumed;
  do {
    assumed = old;
    if (__uint_as_float(assumed) >= val) break;
    old = atomicCAS(ua, assumed, __float_as_uint(val));
  } while (old != assumed);
  return __uint_as_float(old);
}

// Pass 1: segment max of leaky_relu(a_src[src]+a_dst[dst]) over dst.
__global__ void edge_seg_max(const int* __restrict__ ei, int E0, int N, int H,
                             const float* __restrict__ a_src,
                             const float* __restrict__ a_dst,
                             float* __restrict__ m) {
  long long i     = (long long)blockIdx.x * blockDim.x + threadIdx.x;
  long long total = (long long)(E0 + N) * H;
  if (i >= total) return;
  int e  = (int)(i / H);
  int hd = (int)(i % H);
  int s  = edge_src(ei, E0, e);
  int d  = edge_dst(ei, E0, e);
  float a = leaky(a_src[(size_t)s * H + hd] + a_dst[(size_t)d * H + hd]);
  atomicMaxF32(&m[(size_t)d * H + hd], a);
}

// Pass 2: denom[dst,h] = sum exp(alpha - m[dst,h]).
__global__ void edge_exp_sum(const int* __restrict__ ei, int E0, int N, int H,
                             const float* __restrict__ a_src,
                             const float* __restrict__ a_dst,
                             const float* __restrict__ m,
                             float* __restrict__ den) {
  long long i     = (long long)blockIdx.x * blockDim.x + threadIdx.x;
  long long total = (long long)(E0 + N) * H;
  if (i >= total) return;
  int e  = (int)(i / H);
  int hd = (int)(i % H);
  int s  = edge_src(ei, E0, e);
  int d  = edge_dst(ei, E0, e);
  size_t dh = (size_t)d * H + hd;
  float a  = leaky(a_src[(size_t)s * H + hd] + a_dst[dh]);
  float ex = expf(a - m[dh]);
  atomicAdd(&den[dh], ex);     // global_atomic_add_f32 (no return)
}

// Pass 3: out[dst,h,:] += h[src,h,:] * exp(alpha-m)/(den+eps).
__global__ void edge_aggregate(const int* __restrict__ ei, int E0, int N,
                               int H, int C,
                               const float* __restrict__ a_src,
                               const float* __restrict__ a_dst,
                               const float* __restrict__ m,
                               const float* __restrict__ den,
                               const float* __restrict__ h,
                               float* __restrict__ out) {
  long long i     = (long long)blockIdx.x * blockDim.x + threadIdx.x;
  long long total = (long long)(E0 + N) * H;
  if (i >= total) return;
  int e  = (int)(i / H);
  int hd = (int)(i % H);
  int s  = edge_src(ei, E0, e);
  int d  = edge_dst(ei, E0, e);
  size_t dh = (size_t)d * H + hd;
  float a     = leaky(a_src[(size_t)s * H + hd] + a_dst[dh]);
  float alpha = expf(a - m[dh]) / (den[dh] + EPSF);
  const float* hs = h + ((size_t)s * H + hd) * C;
  float*       op = out + (dh)*C;
  for (int c = 0; c < C; ++c) atomicAdd(&op[c], hs[c] * alpha);
}

// acc = elu(acc + bias)  (in place), bias indexed mod HC.
__global__ void bias_elu(float* __restrict__ p, const float* __restrict__ bias,
                         int HC, long long n) {
  long long i      = (long long)blockIdx.x * blockDim.x + threadIdx.x;
  long long stride = (long long)gridDim.x * blockDim.x;
  for (; i < n; i += stride) {
    float v = p[i] + bias[i % HC];
    p[i] = v > 0.f ? v : expm1f(v);
  }
}

// p += bias (mod HC), in place.
__global__ void bias_add(float* __restrict__ p, const float* __restrict__ bias,
                         int HC, long long n) {
  long long i      = (long long)blockIdx.x * blockDim.x + threadIdx.x;
  long long stride = (long long)gridDim.x * blockDim.x;
  for (; i < n; i += stride) p[i] = p[i] + bias[i % HC];
}

// ---------------------------------------------------------------------------
extern "C" void kernel_launch(void* const* d_in, const int* in_sizes, int n_in,
                              void* d_out, int out_size, void* d_ws, size_t ws_size,
                              hipStream_t stream) {
  const float* x    = (const float*)d_in[0];   // [N,128]
  const int*   ei   = (const int*)d_in[1];     // [2,E0]
  const float* W1   = (const float*)d_in[2];   // [128,256]
  const float* as1w = (const float*)d_in[3];   // [8,32]
  const float* ad1w = (const float*)d_in[4];   // [8,32]
  const float* b1   = (const float*)d_in[5];   // [256]
  const float* W2   = (const float*)d_in[6];   // [256,32]
  const float* as2w = (const float*)d_in[7];   // [1,32]
  const float* ad2w = (const float*)d_in[8];   // [1,32]
  const float* b2   = (const float*)d_in[9];   // [32]
  float* out = (float*)d_out;                  // [N,32]

  const int IN_CH = 128, H1 = 8, C1 = 32, HC1 = 256, C2 = 32;
  const int N  = in_sizes[0] / IN_CH;          // 50000
  const int E0 = in_sizes[1] / 2;              // 800000

  // Workspace carve-up (floats).
  float* ws    = (float*)d_ws;
  float* h1    = ws;                              // N*256
  float* acc1  = h1    + (size_t)N * HC1;         // N*256
  float* asrc1 = acc1  + (size_t)N * HC1;         // N*8
  float* adst1 = asrc1 + (size_t)N * H1;          // N*8
  float* m1    = adst1 + (size_t)N * H1;          // N*8
  float* den1  = m1    + (size_t)N * H1;          // N*8
  float* h2    = den1  + (size_t)N * H1;          // N*32
  float* asrc2 = h2    + (size_t)N * C2;          // N
  float* adst2 = asrc2 + (size_t)N;               // N
  float* m2    = adst2 + (size_t)N;               // N
  float* den2  = m2    + (size_t)N;               // N

  const float NEG_BIG = -3.402823466e38f;

  // ---- init accumulators ----
  fill_f32<<<2048, 256, 0, stream>>>(acc1, 0.f, (long long)N * HC1);
  fill_f32<<<2048, 256, 0, stream>>>(out,  0.f, (long long)N * C2);
  fill_f32<<<512, 256, 0, stream>>>(den1, 0.f, (long long)N * H1);
  fill_f32<<<512, 256, 0, stream>>>(m1, NEG_BIG, (long long)N * H1);
  fill_f32<<<256, 256, 0, stream>>>(den2, 0.f, (long long)N);
  fill_f32<<<256, 256, 0, stream>>>(m2, NEG_BIG, (long long)N);

  // ---- Layer 1 ----
  {
    int tiles  = (N / 16) * (HC1 / 16);          // 3125 * 16
    int blocks = (tiles + 7) / 8;                // 8 waves (256 thr) per block
    gemm16_wmma_f32<<<blocks, 256, 0, stream>>>(x, W1, h1, N, IN_CH, HC1);
  }
  att_scores<<<(N * H1 + 255) / 256, 256, 0, stream>>>(h1, as1w, ad1w,
                                                       asrc1, adst1, N, H1, C1);
  {
    long long tot = (long long)(E0 + N) * H1;
    int blocks = (int)((tot + 255) / 256);
    edge_seg_max<<<blocks, 256, 0, stream>>>(ei, E0, N, H1, asrc1, adst1, m1);
    edge_exp_sum<<<blocks, 256, 0, stream>>>(ei, E0, N, H1, asrc1, adst1, m1, den1);
    edge_aggregate<<<blocks, 256, 0, stream>>>(ei, E0, N, H1, C1, asrc1, adst1,
                                               m1, den1, h1, acc1);
  }
  bias_elu<<<2048, 256, 0, stream>>>(acc1, b1, HC1, (long long)N * HC1);

  // ---- Layer 2 ----
  {
    int tiles  = (N / 16) * (C2 / 16);           // 3125 * 2
    int blocks = (tiles + 7) / 8;
    gemm16_wmma_f32<<<blocks, 256, 0, stream>>>(acc1, W2, h2, N, HC1, C2);
  }
  att_scores<<<(N + 255) / 256, 256, 0, stream>>>(h2, as2w, ad2w,
                                                  asrc2, adst2, N, 1, C2);
  {
    long long tot = (long long)(E0 + N);
    int blocks = (int)((tot + 255) / 256);
    edge_seg_max<<<blocks, 256, 0, stream>>>(ei, E0, N, 1, asrc2, adst2, m2);
    edge_exp_sum<<<blocks, 256, 0, stream>>>(ei, E0, N, 1, asrc2, adst2, m2, den2);
    edge_aggregate<<<blocks, 256, 0, stream>>>(ei, E0, N, 1, C2, asrc2, adst2,
                                               m2, den2, h2, out);
  }
  bias_add<<<1024, 256, 0, stream>>>(out, b2, C2, (long long)N * C2);
}